// MambaSOCModel_31404800868989
// MI455X (gfx1250) — compile-verified
//
#include <hip/hip_runtime.h>
#include <hip/hip_bf16.h>

#define D_MODEL 256
#define N_LAYERS 4
#define D_STATE 16
#define D_CONV 4
#define D_INNER 512
#define DT_RANK 16
#define BATCH 8
#define SEQ 2048
#define NTOK (BATCH * SEQ)
#define DBC_W (DT_RANK + 2 * D_STATE) /* 48 */

typedef __attribute__((ext_vector_type(2))) float v2f;
typedef __attribute__((ext_vector_type(4))) float v4f;
typedef __attribute__((ext_vector_type(8))) float v8f;
typedef __attribute__((ext_vector_type(4))) int   v4i;

// ---- CDNA5 async global->LDS path (toolchain-guarded) ---------------------
#if __has_builtin(__builtin_amdgcn_global_load_async_to_lds_b128)
#define ASYNC_OK 1
typedef __attribute__((address_space(1))) v4i* gv4i_p;   // global v4i*
typedef __attribute__((address_space(3))) v4i* lv4i_p;   // LDS v4i*
#if __has_builtin(__builtin_amdgcn_s_wait_asynccnt)
#define WAIT_ASYNC(n) __builtin_amdgcn_s_wait_asynccnt(n)
#else
#define WAIT_ASYNC(n) asm volatile("s_wait_asynccnt %0" ::"i"(n) : "memory")
#endif
#else
#define ASYNC_OK 0
#define WAIT_ASYNC(n)
#endif

// ---------------------------------------------------------------------------
// fp32 WMMA GEMM:  C[M,N] (ldc) = act( A[M,K] (lda) @ W[N,K]^T + bias )
//                  optionally += existing C (residual)
//
// Block = 256 threads = 8 waves. Block tile: 128 rows x (NT*16) cols.
// W panel staged in LDS (double-buffered KC-chunks, async global->LDS b128),
// shared by all 8 waves; each wave holds NT accumulators and reuses its A
// fragment across NT WMMAs. LDS rows padded (+4 floats) for bank-conflict-free
// 16-lane B-fragment reads.
// ACT: 0 = none, 1 = softplus
// ---------------------------------------------------------------------------
template <int K, int NT, int ACT, bool RESID, bool BIAS>
__global__ __launch_bounds__(256) void gemm_wmma_f32(
    const float* __restrict__ A, int lda,
    const float* __restrict__ W,          // N x K row-major
    const float* __restrict__ bias,
    float* __restrict__ C, int ldc,
    int M, int N)
{
    constexpr int KC   = (K < 64) ? K : 64;   // K-chunk staged in LDS
    constexpr int LDW  = KC + 4;              // padded LDS row (bank-stagger)
    constexpr int NROW = NT * 16;             // W rows per block panel
    constexpr int CHUNK_OPS = NROW * (KC / 4);   // b128 transfers per chunk
    constexpr int NPT  = CHUNK_OPS / 256;        // per-thread transfers (exact)

    __shared__ __align__(16) float Wl[2][NROW][LDW];

    const int tid  = threadIdx.x;
    const int wave = tid >> 5;
    const int lane = tid & 31;

    const int nblk = N / NROW;
    const int mb = blockIdx.x / nblk;
    const int nb = blockIdx.x % nblk;
    const int m0 = mb * 128 + wave * 16;
    const int n0 = nb * NROW;

    const int row16 = lane & 15;           // A-frag M row / B-frag N col
    const int kb    = (lane >> 4) << 1;    // K pair base: 0 or 2

    const float* __restrict__ Arow = A + (size_t)(m0 + row16) * lda + kb;

    auto stage = [&](int buf, int kc) {
#pragma unroll
        for (int i = 0; i < NPT; ++i) {
            const int idx = tid + i * 256;
            const int r = idx / (KC / 4);
            const int c = (idx % (KC / 4)) * 4;
#if ASYNC_OK
            __builtin_amdgcn_global_load_async_to_lds_b128(
                (gv4i_p)(W + (size_t)(n0 + r) * K + kc + c),
                (lv4i_p)&Wl[buf][r][c], 0, 0);
#else
            *(v4f*)&Wl[buf][r][c] = *(const v4f*)(W + (size_t)(n0 + r) * K + kc + c);
#endif
        }
    };

    v8f acc[NT];
#pragma unroll
    for (int t = 0; t < NT; ++t) acc[t] = (v8f){};

    stage(0, 0);
    for (int kc = 0, it = 0; kc < K; kc += KC, ++it) {
        const int buf = it & 1;
        if (kc + KC < K) {
            stage(buf ^ 1, kc + KC);   // prefetch next chunk behind compute
            WAIT_ASYNC(NPT);           // current chunk resident (in-order)
        } else {
            WAIT_ASYNC(0);             // last chunk: everything resident
        }
        __syncthreads();
#pragma unroll 4
        for (int k = 0; k < KC; k += 4) {
            v2f a;
            a.x = Arow[kc + k];
            a.y = Arow[kc + k + 1];
#pragma unroll
            for (int t = 0; t < NT; ++t) {
                v2f b;
                b.x = Wl[buf][t * 16 + row16][k + kb];
                b.y = Wl[buf][t * 16 + row16][k + kb + 1];
                acc[t] = __builtin_amdgcn_wmma_f32_16x16x4_f32(
                    /*neg_a=*/false, a, /*neg_b=*/false, b,
                    /*c_mod=*/(short)0, acc[t], /*reuse_a=*/false, /*reuse_b=*/false);
            }
        }
        __syncthreads();
    }

    const int mbase = (lane >> 4) << 3;    // 0 or 8
#pragma unroll
    for (int t = 0; t < NT; ++t) {
        const int n = n0 + t * 16 + row16;
        const float bs = BIAS ? bias[n] : 0.0f;
#pragma unroll
        for (int v = 0; v < 8; ++v) {
            const int m = m0 + mbase + v;
            float val = acc[t][v] + bs;
            if (ACT == 1) val = (val > 20.0f) ? val : log1pf(__expf(val));
            float* cp = &C[(size_t)m * ldc + n];
            if (RESID) val += *cp;
            *cp = val;
        }
    }
}

// ---------------------------------------------------------------------------
// h[tok,e] = x[tok,0]*w_in[e,0] + x[tok,1]*w_in[e,1] + b_in[e]
// ---------------------------------------------------------------------------
__global__ __launch_bounds__(256) void embed_kernel(
    const float* __restrict__ x, const float* __restrict__ w_in,
    const float* __restrict__ b_in, float* __restrict__ h)
{
    int idx = blockIdx.x * 256 + threadIdx.x;   // NTOK * 256
    int e = idx & 255;
    int tok = idx >> 8;
    h[idx] = x[tok * 2] * w_in[e * 2] + x[tok * 2 + 1] * w_in[e * 2 + 1] + b_in[e];
}

// ---------------------------------------------------------------------------
// RMSNorm: one 256-thread block per token
// ---------------------------------------------------------------------------
__global__ __launch_bounds__(256) void rmsnorm_kernel(
    const float* __restrict__ h, const float* __restrict__ w,
    float* __restrict__ xn)
{
    const int tok = blockIdx.x;
    const int e = threadIdx.x;
    float v = h[(size_t)tok * D_MODEL + e];
    __shared__ float red[256];
    red[e] = v * v;
    __syncthreads();
    for (int s = 128; s > 0; s >>= 1) {
        if (e < s) red[e] += red[e + s];
        __syncthreads();
    }
    float inv = rsqrtf(red[0] * (1.0f / D_MODEL) + 1e-5f);
    xn[(size_t)tok * D_MODEL + e] = v * inv * w[e];
}

// ---------------------------------------------------------------------------
// Causal depthwise conv (width 4) over u = xz[:, :512], + bias, then SiLU.
// ---------------------------------------------------------------------------
__global__ __launch_bounds__(256) void conv_silu_kernel(
    const float* __restrict__ xz, const float* __restrict__ cw,
    const float* __restrict__ cb, float* __restrict__ uc)
{
    int idx = blockIdx.x * 256 + threadIdx.x;   // NTOK * 512
    int d = idx & (D_INNER - 1);
    int tok = idx >> 9;
    int l = tok & (SEQ - 1);
    float acc = cb[d];
#pragma unroll
    for (int j = 0; j < D_CONV; ++j) {
        int ls = l - (D_CONV - 1) + j;
        if (ls >= 0)
            acc += xz[(size_t)(tok - (D_CONV - 1) + j) * (2 * D_INNER) + d] * cw[d * D_CONV + j];
    }
    uc[idx] = acc / (1.0f + __expf(-acc));      // silu
}

// ---------------------------------------------------------------------------
// Selective scan + output gate. One thread per (batch, channel); 16 states in
// registers; per-step B/C (128 contiguous bytes) staged to LDS via async
// global->LDS b128 (lanes 0-7 of wave 0), shared by 256 channels.
// g[tok,d] = (sum_n h_n*C_n + u*Dp) * silu(z)
// ---------------------------------------------------------------------------
__global__ __launch_bounds__(256) void scan_gate_kernel(
    const float* __restrict__ uc, const float* __restrict__ delta,
    const float* __restrict__ dbc, const float* __restrict__ xz,
    const float* __restrict__ A_log, const float* __restrict__ D_skip,
    float* __restrict__ g)
{
    const int b    = blockIdx.x >> 1;
    const int half = blockIdx.x & 1;
    const int d    = half * 256 + threadIdx.x;

    float A[D_STATE];
#pragma unroll
    for (int n = 0; n < D_STATE; ++n)
        A[n] = -__expf(A_log[(size_t)d * D_STATE + n]);
    const float Dp = D_skip[d];

    float hs[D_STATE];
#pragma unroll
    for (int n = 0; n < D_STATE; ++n) hs[n] = 0.0f;

    __shared__ __align__(16) float BC[2 * D_STATE];

    for (int l = 0; l < SEQ; ++l) {
        const int tok = b * SEQ + l;
#if ASYNC_OK
        if (threadIdx.x < 8)
            __builtin_amdgcn_global_load_async_to_lds_b128(
                (gv4i_p)(dbc + (size_t)tok * DBC_W + DT_RANK + threadIdx.x * 4),
                (lv4i_p)&BC[threadIdx.x * 4], 0, 0);
        WAIT_ASYNC(0);
#else
        if (threadIdx.x < 2 * D_STATE)
            BC[threadIdx.x] = dbc[(size_t)tok * DBC_W + DT_RANK + threadIdx.x];
#endif
        __syncthreads();

        const float dt = delta[(size_t)tok * D_INNER + d];
        const float u  = uc[(size_t)tok * D_INNER + d];
        const float du = dt * u;
        float acc = 0.0f;
#pragma unroll
        for (int n = 0; n < D_STATE; ++n) {
            float dA = __expf(dt * A[n]);
            hs[n] = dA * hs[n] + du * BC[n];
            acc += hs[n] * BC[D_STATE + n];
        }
        const float y = acc + u * Dp;
        const float z = xz[(size_t)tok * (2 * D_INNER) + D_INNER + d];
        const float sz = z / (1.0f + __expf(-z));
        g[(size_t)tok * D_INNER + d] = y * sz;
        __syncthreads();
    }
}

// ---------------------------------------------------------------------------
// Head: out[tok] = dot(h[tok,:], w_head) + b_head   (one wave per token)
// ---------------------------------------------------------------------------
__global__ __launch_bounds__(256) void head_kernel(
    const float* __restrict__ h, const float* __restrict__ wh,
    const float* __restrict__ bh, float* __restrict__ out)
{
    const int wave = threadIdx.x >> 5;
    const int lane = threadIdx.x & 31;
    const int tok = blockIdx.x * 8 + wave;
    float acc = 0.0f;
    for (int e = lane; e < D_MODEL; e += 32)
        acc += h[(size_t)tok * D_MODEL + e] * wh[e];
    for (int off = 16; off > 0; off >>= 1)
        acc += __shfl_down(acc, off, 32);
    if (lane == 0) out[tok] = acc + bh[0];
}

// ---------------------------------------------------------------------------
extern "C" void kernel_launch(void* const* d_in, const int* in_sizes, int n_in,
                              void* d_out, int out_size, void* d_ws, size_t ws_size,
                              hipStream_t stream)
{
    const float* x         = (const float*)d_in[0];
    const float* w_in      = (const float*)d_in[1];
    const float* b_in      = (const float*)d_in[2];
    const float* norm_w    = (const float*)d_in[3];
    const float* in_proj_w = (const float*)d_in[4];
    const float* conv_w    = (const float*)d_in[5];
    const float* conv_b    = (const float*)d_in[6];
    const float* x_proj_w  = (const float*)d_in[7];
    const float* dt_proj_w = (const float*)d_in[8];
    const float* dt_proj_b = (const float*)d_in[9];
    const float* A_log     = (const float*)d_in[10];
    const float* D_skip    = (const float*)d_in[11];
    const float* out_proj_w= (const float*)d_in[12];
    const float* w_head    = (const float*)d_in[13];
    const float* b_head    = (const float*)d_in[14];
    float* out = (float*)d_out;

    // workspace layout (floats)
    float* ws = (float*)d_ws;
    float* h     = ws;                         // NTOK*256
    float* xn    = h     + (size_t)NTOK * D_MODEL;
    float* xz    = xn    + (size_t)NTOK * D_MODEL;      // NTOK*1024
    float* uc    = xz    + (size_t)NTOK * 2 * D_INNER;  // NTOK*512
    float* dbc   = uc    + (size_t)NTOK * D_INNER;      // NTOK*48
    float* delta = dbc   + (size_t)NTOK * DBC_W;        // NTOK*512
    float* g     = delta + (size_t)NTOK * D_INNER;      // NTOK*512

    // embed: h = x @ w_in.T + b_in
    embed_kernel<<<(NTOK * D_MODEL) / 256, 256, 0, stream>>>(x, w_in, b_in, h);

    for (int layer = 0; layer < N_LAYERS; ++layer) {
        const float* nw  = norm_w    + (size_t)layer * D_MODEL;
        const float* ipw = in_proj_w + (size_t)layer * 2 * D_INNER * D_MODEL;
        const float* cw  = conv_w    + (size_t)layer * D_INNER * D_CONV;
        const float* cb  = conv_b    + (size_t)layer * D_INNER;
        const float* xpw = x_proj_w  + (size_t)layer * DBC_W * D_INNER;
        const float* dtw = dt_proj_w + (size_t)layer * D_INNER * DT_RANK;
        const float* dtb = dt_proj_b + (size_t)layer * D_INNER;
        const float* Al  = A_log     + (size_t)layer * D_INNER * D_STATE;
        const float* Dsk = D_skip    + (size_t)layer * D_INNER;
        const float* opw = out_proj_w+ (size_t)layer * D_MODEL * D_INNER;

        // xn = rmsnorm(h) * norm_w
        rmsnorm_kernel<<<NTOK, 256, 0, stream>>>(h, nw, xn);

        // xz = xn @ in_proj_w.T            M=16384 N=1024 K=256
        gemm_wmma_f32<256, 4, 0, false, false>
            <<<(NTOK / 128) * (1024 / 64), 256, 0, stream>>>(
                xn, D_MODEL, ipw, nullptr, xz, 2 * D_INNER, NTOK, 2 * D_INNER);

        // uc = silu(causal_dwconv(u) + conv_b)
        conv_silu_kernel<<<(NTOK * D_INNER) / 256, 256, 0, stream>>>(xz, cw, cb, uc);

        // dbc = uc @ x_proj_w.T            M=16384 N=48 K=512
        gemm_wmma_f32<512, 3, 0, false, false>
            <<<(NTOK / 128) * (48 / 48), 256, 0, stream>>>(
                uc, D_INNER, xpw, nullptr, dbc, DBC_W, NTOK, DBC_W);

        // delta = softplus(dbc[:, :16] @ dt_proj_w.T + dt_proj_b)  M=16384 N=512 K=16
        gemm_wmma_f32<16, 4, 1, false, true>
            <<<(NTOK / 128) * (512 / 64), 256, 0, stream>>>(
                dbc, DBC_W, dtw, dtb, delta, D_INNER, NTOK, D_INNER);

        // selective scan + silu(z) gate -> g
        scan_gate_kernel<<<2 * BATCH, 256, 0, stream>>>(
            uc, delta, dbc, xz, Al, Dsk, g);

        // h += g @ out_proj_w.T            M=16384 N=256 K=512  (residual)
        gemm_wmma_f32<512, 4, 0, true, false>
            <<<(NTOK / 128) * (256 / 64), 256, 0, stream>>>(
                g, D_INNER, opw, nullptr, h, D_MODEL, NTOK, D_MODEL);
    }

    // out = h @ w_head.T + b_head
    head_kernel<<<NTOK / 8, 256, 0, stream>>>(h, w_head, b_head, out);
}